// MultiHeadAttention_46926812676474
// MI455X (gfx1250) — compile-verified
//
#include <hip/hip_runtime.h>
#include <math.h>
#include <stdint.h>

// ---------------------------------------------------------------------------
// MultiHeadAttention with depth bias — MI455X (gfx1250, wave32, WMMA f32 path)
// Shapes: B=8, S=768 (V=384 + A=384), Hd=512, H=8, d=64.
// HBM-bound on att_score/att_map (302 MB written + 151 MB re-read).
// WMMA f32 16x16x4 for all GEMMs; async global->LDS staging of shared A tiles;
// 16x64 wave tiles (4 accumulators) to amortize A operands 4x per WMMA.
// ---------------------------------------------------------------------------

typedef __attribute__((ext_vector_type(2))) float v2f;
typedef __attribute__((ext_vector_type(8))) float v8f;

#define HN     8
#define HD     512
#define DH     64
#define SEQ    768
#define VN     384
#define AN     384
#define BN     8
#define REP    4
#define BBC    0.6f
#define NEGV   (-1000000000.0f)

#define LDA_GEMM  516   // 512 + 4 pad: LDS bank stride 4 -> conflict-free halves
#define LDA_SCORE 68    // 64 + 4 pad

static __device__ __forceinline__ v8f wmma_f32_k4(v2f a, v2f b, v8f c) {
  // V_WMMA_F32_16X16X4_F32 : D = A(16x4) x B(4x16) + C(16x16)
  return __builtin_amdgcn_wmma_f32_16x16x4_f32(
      false, a, false, b, (short)0, c, false, false);
}

// Async global->LDS copy, 16B per lane, tracked by ASYNCcnt (CDNA5 path).
static __device__ __forceinline__ void async_ld_b128(uint32_t lds_off,
                                                     const float* gaddr) {
  asm volatile("global_load_async_to_lds_b128 %0, %1, off"
               :: "v"(lds_off), "v"(gaddr) : "memory");
}
static __device__ __forceinline__ void wait_async0() {
  asm volatile("s_wait_asynccnt 0x0" ::: "memory");
}
static __device__ __forceinline__ uint32_t lds_off_of(const float* p) {
  return (uint32_t)(uintptr_t)p;  // flat->LDS keeps addr[31:0]
}

// --------------------------- block reductions ------------------------------
static __device__ __forceinline__ float blk_max(float v, float* red, int tid) {
  red[tid] = v; __syncthreads();
  #pragma unroll
  for (int off = 128; off > 0; off >>= 1) {
    if (tid < off) red[tid] = fmaxf(red[tid], red[tid + off]);
    __syncthreads();
  }
  float r = red[0]; __syncthreads();
  return r;
}
static __device__ __forceinline__ float blk_sum(float v, float* red, int tid) {
  red[tid] = v; __syncthreads();
  #pragma unroll
  for (int off = 128; off > 0; off >>= 1) {
    if (tid < off) red[tid] += red[tid + off];
    __syncthreads();
  }
  float r = red[0]; __syncthreads();
  return r;
}

// ---------------------------------------------------------------------------
// 6144x512 @ 512x512 + bias. Block = one 16-row M tile, 8 waves x 64 N cols.
// A tile (16x512, 32KB) staged to LDS via async b128; each wave runs 4 WMMAs
// per K-step (16x64 tile) reusing one LDS A operand. Grid: 384 x 256.
// ---------------------------------------------------------------------------
__global__ void gemm512_bias(const float* __restrict__ X,
                             const float* __restrict__ W,
                             const float* __restrict__ bias,
                             float* __restrict__ Out,
                             int headmajor) {
  __shared__ float tileA[16 * LDA_GEMM];
  const int tid   = threadIdx.x;
  const int lane  = tid & 31;
  const int w     = tid >> 5;          // wave 0..7 -> N group
  const int tm    = blockIdx.x;        // 0..383
  const int l15   = lane & 15;
  const int khalf = lane >> 4;
  const int m0    = tm * 16;
  const int n0    = w * 64;

  // Stage A tile: 16 rows x 512 floats = 2048 vec4; 8 per thread.
  {
    const int vr = tid;                          // 256 threads
    #pragma unroll
    for (int it = 0; it < 8; ++it) {
      const int v    = vr + 256 * it;            // 0..2047
      const int row  = v >> 7;                   // /128 vec4 per row
      const int col4 = v & 127;
      async_ld_b128(lds_off_of(&tileA[row * LDA_GEMM + col4 * 4]),
                    X + (size_t)(m0 + row) * HD + col4 * 4);
    }
  }
  wait_async0();
  __syncthreads();

  v8f acc[4] = {};
  #pragma unroll 4
  for (int kk = 0; kk < HD; kk += 4) {
    const int kr = kk + 2 * khalf;
    const v2f a = *(const v2f*)&tileA[l15 * LDA_GEMM + kr];  // ds_load_b64
    #pragma unroll
    for (int t = 0; t < 4; ++t) {
      const int nc = n0 + t * 16 + l15;
      v2f b;
      b.x = W[(size_t)kr * HD + nc];
      b.y = W[(size_t)(kr + 1) * HD + nc];
      acc[t] = wmma_f32_k4(a, b, acc[t]);
    }
  }

  #pragma unroll
  for (int t = 0; t < 4; ++t) {
    const int nc = n0 + t * 16 + l15;
    const float bn = bias[nc];
    #pragma unroll
    for (int i = 0; i < 8; ++i) {
      const int m   = m0 + i + 8 * khalf;
      const float v = acc[t][i] + bn;
      if (headmajor) {
        const int bb = m / SEQ, s = m % SEQ;
        const int hh = nc / DH, dd = nc % DH;
        Out[(((size_t)(bb * HN + hh)) * SEQ + s) * DH + dd] = v;
      } else {
        Out[(size_t)m * HD + nc] = v;
      }
    }
  }
}

// ---------------------------------------------------------------------------
// Scores: per (b,h) X = scale * Qh(768x64) @ Kh^T(64x768).
// Block = one 16-row Q tile (staged 16x64 via async) shared by 8 waves, each
// wave one 16x16 K-column tile (B loads contiguous since B[k][n] = Kh[n][k]).
// Grid: (288, 64) x 256; blockIdx.x = tm*6 + tgroup.
// ---------------------------------------------------------------------------
__global__ void scores_wmma(const float* __restrict__ qh,
                            const float* __restrict__ kh,
                            float* __restrict__ xout, float scale) {
  __shared__ float tileA[16 * LDA_SCORE];
  const int bh    = blockIdx.y;
  const int tid   = threadIdx.x;
  const int lane  = tid & 31;
  const int w     = tid >> 5;
  const int tm    = blockIdx.x / 6;            // 0..47
  const int tn    = (blockIdx.x % 6) * 8 + w;  // 0..47
  const int l15   = lane & 15;
  const int khalf = lane >> 4;
  const float* Q  = qh + (size_t)bh * SEQ * DH;
  const float* Kh = kh + (size_t)bh * SEQ * DH;

  // Stage Q tile: 16 x 64 floats = 256 vec4; one per thread.
  {
    const int row  = tid >> 4;   // 0..15
    const int col4 = tid & 15;   // 0..15
    async_ld_b128(lds_off_of(&tileA[row * LDA_SCORE + col4 * 4]),
                  Q + (size_t)(tm * 16 + row) * DH + col4 * 4);
  }
  wait_async0();
  __syncthreads();

  const int kc = tn * 16 + l15;
  v8f acc = {};
  #pragma unroll
  for (int kk = 0; kk < DH; kk += 4) {
    const int kr = kk + 2 * khalf;
    const v2f a = *(const v2f*)&tileA[l15 * LDA_SCORE + kr];
    const v2f b = *(const v2f*)(Kh + (size_t)kc * DH + kr);
    acc = wmma_f32_k4(a, b, acc);
  }
  float* Xo = xout + (size_t)bh * SEQ * SEQ;
  #pragma unroll
  for (int i = 0; i < 8; ++i) {
    const int m = tm * 16 + i + 8 * khalf;
    Xo[(size_t)m * SEQ + tn * 16 + l15] = acc[i] * scale;
  }
}

// ---------------------------------------------------------------------------
// Fused depth-bias + masked softmax + av-softmax + md accumulation.
// One block (256 threads) per row (b,h,q); 3 elements per thread (S=768).
// ---------------------------------------------------------------------------
__global__ void bias_softmax(float* __restrict__ att,
                             float* __restrict__ amap,
                             const float* __restrict__ mask,
                             const float* __restrict__ depth_value,
                             const float* __restrict__ mean_depth_value,
                             float* __restrict__ md_acc) {
  __shared__ float red[256];
  const int row = blockIdx.x;
  const int b   = row / (HN * SEQ);
  const int q   = row % SEQ;
  const int tid = threadIdx.x;
  const int b0  = b / REP;

  float* arow = att  + (size_t)row * SEQ;
  float* prow = amap + (size_t)row * SEQ;

  const float mq   = mask[b * SEQ + q];
  const float mdva = (q >= VN) ? mean_depth_value[b * AN + (q - VN)] : 0.f;
  const float dvq  = (q <  VN) ? depth_value[b0 * VN + q] : 0.f;

  float xv[3], sv[3], m4[3];
  #pragma unroll
  for (int j = 0; j < 3; ++j) {
    const int k = tid + 256 * j;
    const float x  = arow[k];
    const float mk = mask[b * SEQ + k];
    xv[j] = x;
    m4[j] = mq * mk;
    float biasv = 0.f;
    if (k < VN) {
      const float dvk = depth_value[b0 * VN + k];
      if (q < VN) {  // vv block: |x| * (e^{-|dv_q - dv_k|} - BB) * (q != k)
        biasv = (q == k) ? 0.f
                         : fabsf(x) * (__expf(-fabsf(dvq - dvk)) - BBC);
      } else {       // av block: |x| * (e^{-|mdv_a - dv_k|} - BB)
        biasv = fabsf(x) * (__expf(-fabsf(mdva - dvk)) - BBC);
      }
    }
    sv[j] = (m4[j] > 0.f) ? (x + biasv) : NEGV;
  }

  // full-row masked softmax:  p = e*m / (sum(e*m) + 1e-13*sum(e))
  float lmax = fmaxf(fmaxf(sv[0], sv[1]), sv[2]);
  const float rmax = blk_max(lmax, red, tid);
  float e[3], ls_e = 0.f, ls_em = 0.f;
  #pragma unroll
  for (int j = 0; j < 3; ++j) {
    e[j] = __expf(sv[j] - rmax);
    ls_e  += e[j];
    ls_em += e[j] * m4[j];
  }
  const float s_e  = blk_sum(ls_e,  red, tid);
  const float s_em = blk_sum(ls_em, red, tid);
  const float inv  = 1.0f / (s_em + 1e-13f * s_e);
  #pragma unroll
  for (int j = 0; j < 3; ++j) {
    const int k = tid + 256 * j;
    arow[k] = sv[j];
    prow[k] = e[j] * m4[j] * inv;
  }

  // av sub-softmax (raw x over k < V) -> md accumulation
  if (q >= VN) {
    float sav[3];
    #pragma unroll
    for (int j = 0; j < 3; ++j) {
      const int k = tid + 256 * j;
      sav[j] = (k < VN && m4[j] > 0.f) ? xv[j] : NEGV;
    }
    float lam = fmaxf(fmaxf(sav[0], sav[1]), sav[2]);
    const float avmax = blk_max(lam, red, tid);
    float ea[3], la_e = 0.f, la_em = 0.f;
    #pragma unroll
    for (int j = 0; j < 3; ++j) {
      const int k = tid + 256 * j;
      ea[j] = (k < VN) ? __expf(sav[j] - avmax) : 0.f;
      la_e  += ea[j];
      la_em += ea[j] * m4[j];
    }
    const float a_e  = blk_sum(la_e,  red, tid);
    const float a_em = blk_sum(la_em, red, tid);
    const float ainv = 1.0f / (a_em + 1e-13f * a_e);
    float ldot = 0.f;
    #pragma unroll
    for (int j = 0; j < 3; ++j) {
      const int k = tid + 256 * j;
      if (k < VN) ldot += (ea[j] * m4[j] * ainv) * depth_value[b0 * VN + k];
    }
    const float dot = blk_sum(ldot, red, tid);
    if (tid == 0) atomicAdd(&md_acc[b], dot);
  }
}

// ---------------------------------------------------------------------------
// Context: per (b,h) Ctx = att_map(768x768) @ Vh(768x64), written (B,S,Hd).
// Wave computes full 16x64 tile (4 accumulators): one contiguous A load feeds
// 4 WMMAs per K-step; cuts att_map L2 read amplification 4x.
// Grid: (6, 64) x 256 -> 48 waves per (b,h).
// ---------------------------------------------------------------------------
__global__ void context_wmma(const float* __restrict__ amap,
                             const float* __restrict__ vh,
                             float* __restrict__ ctx) {
  const int bh    = blockIdx.y;
  const int bg    = bh / HN, hh = bh % HN;
  const int lane  = threadIdx.x & 31;
  const int tm    = blockIdx.x * 8 + (threadIdx.x >> 5);  // 0..47
  const int l15   = lane & 15;
  const int khalf = lane >> 4;
  const float* P  = amap + (size_t)bh * SEQ * SEQ;
  const float* Vh = vh   + (size_t)bh * SEQ * DH;
  const int qr = tm * 16 + l15;

  v8f acc[4] = {};
  #pragma unroll 4
  for (int kk = 0; kk < SEQ; kk += 4) {
    const int kr = kk + 2 * khalf;
    const v2f a = *(const v2f*)(P + (size_t)qr * SEQ + kr);
    #pragma unroll
    for (int t = 0; t < 4; ++t) {
      const int nc = t * 16 + l15;
      v2f b;
      b.x = Vh[(size_t)kr * DH + nc];
      b.y = Vh[(size_t)(kr + 1) * DH + nc];
      acc[t] = wmma_f32_k4(a, b, acc[t]);
    }
  }
  #pragma unroll
  for (int t = 0; t < 4; ++t) {
    #pragma unroll
    for (int i = 0; i < 8; ++i) {
      const int qq = tm * 16 + i + 8 * khalf;
      ctx[((size_t)bg * SEQ + qq) * HD + hh * DH + t * 16 + l15] = acc[t][i];
    }
  }
}

// ---------------------------------------------------------------------------
// md finalize: md[b, :] = md_acc[b] / (H * sum_a va_a_mask[b,a])
// ---------------------------------------------------------------------------
__global__ void md_final(const float* __restrict__ md_acc,
                         const float* __restrict__ va_a_mask,
                         float* __restrict__ md_out) {
  __shared__ float red[256];
  const int b = blockIdx.x;
  const int tid = threadIdx.x;
  float c = 0.f;
  for (int a = tid; a < AN; a += 256) c += va_a_mask[b * AN + a];
  const float count = blk_sum(c, red, tid);
  const float val = md_acc[b] / ((float)HN * count);
  for (int a = tid; a < AN; a += 256) md_out[b * AN + a] = val;
}

// ---------------------------------------------------------------------------
extern "C" void kernel_launch(void* const* d_in, const int* in_sizes, int n_in,
                              void* d_out, int out_size, void* d_ws, size_t ws_size,
                              hipStream_t stream) {
  const float* q    = (const float*)d_in[0];
  const float* k    = (const float*)d_in[1];
  const float* v    = (const float*)d_in[2];
  const float* mask = (const float*)d_in[3];
  const float* depth_value      = (const float*)d_in[4];
  // d_in[5] coord_3d  : unused by reference
  // d_in[6] va_v_mask : unused by reference
  const float* va_a_mask        = (const float*)d_in[7];
  const float* mean_depth_value = (const float*)d_in[8];
  const float* Wq = (const float*)d_in[9];   const float* bq = (const float*)d_in[10];
  const float* Wk = (const float*)d_in[11];  const float* bk = (const float*)d_in[12];
  const float* Wv = (const float*)d_in[13];  const float* bv = (const float*)d_in[14];
  const float* Wo = (const float*)d_in[15];  const float* bo = (const float*)d_in[16];

  // d_out layout: out | att_score | att_map | md
  float* out       = (float*)d_out;
  float* att_score = out + (size_t)BN * SEQ * HD;
  float* att_map   = att_score + (size_t)BN * HN * SEQ * SEQ;
  float* md_out    = att_map   + (size_t)BN * HN * SEQ * SEQ;

  // workspace: qh | kh | vh | md_acc   (ctx reuses qh after scores)
  float* qh     = (float*)d_ws;
  float* kh     = qh + (size_t)BN * SEQ * HD;
  float* vh     = kh + (size_t)BN * SEQ * HD;
  float* md_acc = vh + (size_t)BN * SEQ * HD;
  float* ctx    = qh;

  hipMemsetAsync(md_acc, 0, BN * sizeof(float), stream);

  const dim3 blk(256);
  // QKV projections -> head-major [b,h,s,d]  (A tile async-staged in LDS)
  gemm512_bias<<<384, blk, 0, stream>>>(q, Wq, bq, qh, 1);
  gemm512_bias<<<384, blk, 0, stream>>>(k, Wk, bk, kh, 1);
  gemm512_bias<<<384, blk, 0, stream>>>(v, Wv, bv, vh, 1);
  // raw logits into att_score slot (Q tile async-staged in LDS)
  scores_wmma<<<dim3(288, 64), blk, 0, stream>>>(qh, kh, att_score, 0.125f);
  // fused bias + masks + softmaxes + md accumulation
  bias_softmax<<<BN * HN * SEQ, blk, 0, stream>>>(att_score, att_map, mask,
                                                  depth_value, mean_depth_value,
                                                  md_acc);
  // PV -> (B,S,Hd) context (reuses qh workspace)
  context_wmma<<<dim3(6, 64), blk, 0, stream>>>(att_map, vh, ctx);
  // output projection
  gemm512_bias<<<384, blk, 0, stream>>>(ctx, Wo, bo, out, 0);
  // md broadcast
  md_final<<<BN, blk, 0, stream>>>(md_acc, va_a_mask, md_out);
}